// GCNModel_39848706573592
// MI455X (gfx1250) — compile-verified
//
#include <hip/hip_runtime.h>
#include <hip/hip_bf16.h>
#include <math.h>

typedef __attribute__((ext_vector_type(2))) float v2f;
typedef __attribute__((ext_vector_type(8))) float v8f;

#define TPB 256

// ---------------- degree / symmetric norm ----------------
__global__ void k_set1(float* deg, int n) {
    int i = blockIdx.x * blockDim.x + threadIdx.x;
    if (i < n) deg[i] = 1.0f;  // self-loop contributes 1 to every node's in-degree
}

__global__ void k_degree(const long long* __restrict__ dst, float* __restrict__ deg, int E) {
    int e = blockIdx.x * blockDim.x + threadIdx.x;
    if (e < E) atomicAdd(&deg[(int)dst[e]], 1.0f);
}

__global__ void k_rsqrt(float* deg, int n) {
    int i = blockIdx.x * blockDim.x + threadIdx.x;
    if (i < n) { float d = deg[i]; deg[i] = (d > 0.0f) ? rsqrtf(d) : 0.0f; }
}

// ---------------- fp32 WMMA GEMM: C[M,Nb] = A[M,K] x B[K,Nb] ----------------
// B is staged zero-padded into LDS ([K][Npad], Npad = nTilesN*16) so the inner
// loop is branch-free: 1x global_load_b64 (A) + 2x ds_load_b32 (B) + WMMA.
// One wave computes one 16x16 output tile via V_WMMA_F32_16X16X4_F32.
__global__ void __launch_bounds__(TPB)
k_gemm_wmma_f32(const float* __restrict__ A, const float* __restrict__ B,
                float* __restrict__ C, int K, int Nb, int Npad, int nTilesN, int nTiles)
{
    __shared__ float bsh[16384];   // 64 KB max: layer1 uses 256*64 exactly, layer2 64*48

    // cooperative zero-padded fill of B into LDS (all waves participate, then barrier)
    int nelem = K * Npad;
    for (int idx = threadIdx.x; idx < nelem; idx += TPB) {
        int k = idx / Npad;
        int c = idx - k * Npad;
        bsh[idx] = (c < Nb) ? B[k * Nb + c] : 0.0f;
    }
    __syncthreads();

    int tile = blockIdx.x * (TPB / 32) + (threadIdx.x >> 5);
    if (tile >= nTiles) return;                 // wave-uniform: EXEC all-1s for WMMA
    int tm   = tile / nTilesN;
    int tn   = tile - tm * nTilesN;
    int lane = threadIdx.x & 31;
    int half = lane >> 4;                       // 0: lanes 0-15, 1: lanes 16-31
    int l16  = lane & 15;
    int col  = tn * 16 + l16;

    const float* arow = A + (size_t)(tm * 16 + l16) * K;   // A row (M = l16)
    const float* brow = bsh + col;
    v8f acc = {};
    #pragma unroll 4
    for (int k = 0; k < K; k += 4) {
        int kk = k + half * 2;                  // A: lanes 0-15 -> K=k,k+1 ; 16-31 -> k+2,k+3
        v2f a = *(const v2f*)(arow + kk);       // contiguous, 8B-aligned
        v2f b;
        b.x = brow[kk * Npad];                  // B row kk   (VGPR0: K=k | k+2 per lane half)
        b.y = brow[(kk + 1) * Npad];            // B row kk+1 (VGPR1: K=k+1 | k+3)
        acc = __builtin_amdgcn_wmma_f32_16x16x4_f32(false, a, false, b, (short)0, acc,
                                                    false, false);
    }
    if (col < Nb) {
        int rowbase = tm * 16 + half * 8;       // D: VGPR r -> M=r (lanes 0-15), M=r+8 (16-31)
        #pragma unroll
        for (int r = 0; r < 8; ++r)
            C[(size_t)(rowbase + r) * Nb + col] = acc[r];
    }
}

// ---------------- out[i,c] = bias[c] + h[i,c]*dinv[i]^2  (self-loop term) ----------------
__global__ void k_init_out(const float* __restrict__ h, const float* __restrict__ dinv,
                           const float* __restrict__ bias, float* __restrict__ out,
                           int n, int F)
{
    long long idx = (long long)blockIdx.x * blockDim.x + threadIdx.x;
    if (idx >= (long long)n * F) return;
    int i = (int)(idx / F);
    int c = (int)(idx - (long long)i * F);
    float di = dinv[i];
    out[idx] = bias[c] + h[idx] * di * di;
}

// ---------------- edge scatter-add: out[dst] += h[src] * dinv[src]*dinv[dst] ----------------
// groups = F/4; each thread handles 4 consecutive features of one edge (L2-resident atomics).
__global__ void k_aggregate(const float* __restrict__ h, const long long* __restrict__ src,
                            const long long* __restrict__ dst, const float* __restrict__ dinv,
                            float* __restrict__ out, int E, int F, int groups)
{
    long long idx = (long long)blockIdx.x * blockDim.x + threadIdx.x;
    if (idx >= (long long)E * groups) return;
    int e = (int)(idx / groups);
    int g = (int)(idx - (long long)e * groups);
    int s = (int)src[e];
    int d = (int)dst[e];
    float norm = dinv[s] * dinv[d];
    const float* hs = h + (size_t)s * F + g * 4;
    float* od       = out + (size_t)d * F + g * 4;
    #pragma unroll
    for (int j = 0; j < 4; ++j)
        atomicAdd(&od[j], hs[j] * norm);
}

__global__ void k_relu(float* x, long long n) {
    long long i = (long long)blockIdx.x * blockDim.x + threadIdx.x;
    if (i < n) x[i] = fmaxf(x[i], 0.0f);
}

// ---------------- row-wise log_softmax over 40 classes, one wave per row ----------------
__global__ void k_logsoftmax(float* __restrict__ out, int n)
{
    int wave = (int)((blockIdx.x * (unsigned)blockDim.x + threadIdx.x) >> 5);
    int lane = threadIdx.x & 31;
    if (wave >= n) return;
    float* row = out + (size_t)wave * 40;
    float v0 = row[lane];                                   // lane < 32 < 40 always valid
    float v1 = (lane + 32 < 40) ? row[lane + 32] : -INFINITY;
    float m = fmaxf(v0, v1);
    #pragma unroll
    for (int off = 16; off > 0; off >>= 1) m = fmaxf(m, __shfl_xor(m, off, 32));
    float s = expf(v0 - m) + ((lane + 32 < 40) ? expf(v1 - m) : 0.0f);
    #pragma unroll
    for (int off = 16; off > 0; off >>= 1) s += __shfl_xor(s, off, 32);
    float lse = m + logf(s);
    row[lane] = v0 - lse;
    if (lane + 32 < 40) row[lane + 32] = v1 - lse;
}

extern "C" void kernel_launch(void* const* d_in, const int* in_sizes, int n_in,
                              void* d_out, int out_size, void* d_ws, size_t ws_size,
                              hipStream_t stream)
{
    const float*     x   = (const float*)d_in[0];
    const long long* ei  = (const long long*)d_in[1];
    const float*     W1  = (const float*)d_in[2];
    const float*     b1  = (const float*)d_in[3];
    const float*     W2  = (const float*)d_in[4];
    const float*     b2  = (const float*)d_in[5];
    float*           out = (float*)d_out;
    (void)n_in; (void)out_size; (void)ws_size;

    int n = in_sizes[0] / 256;        // 100000 nodes
    int E = in_sizes[1] / 2;          // 3200000 edges
    const long long* src = ei;        // edge_index[0]
    const long long* dst = ei + E;    // edge_index[1]

    // workspace carve-out (~68 MB)
    char* ws = (char*)d_ws;
    size_t off = 0;
    auto carve = [&](size_t bytes) {
        char* p = ws + off;
        off = (off + bytes + 255) & ~(size_t)255;
        return p;
    };
    float* deg  = (float*)carve((size_t)n * 4);          // degree -> dinv (in place)
    float* h1   = (float*)carve((size_t)n * 64 * 4);     // x @ W1
    float* out1 = (float*)carve((size_t)n * 64 * 4);     // aggregated + relu
    float* h2   = (float*)carve((size_t)n * 40 * 4);     // relu(out1) @ W2

    auto blocks = [](long long t) { return (int)((t + TPB - 1) / TPB); };

    // degree / normalization (shared by both layers)
    k_set1  <<<blocks(n), TPB, 0, stream>>>(deg, n);
    k_degree<<<blocks(E), TPB, 0, stream>>>(dst, deg, E);
    k_rsqrt <<<blocks(n), TPB, 0, stream>>>(deg, n);

    int mt = n / 16;                  // 6250 row tiles (exact)

    // ---- layer 1: h1 = x @ W1 ----
    int t1 = mt * 4;                  // N=64 -> 4 col tiles, Npad=64
    k_gemm_wmma_f32<<<(t1 + 7) / 8, TPB, 0, stream>>>(x, W1, h1, 256, 64, 64, 4, t1);
    k_init_out <<<blocks((long long)n * 64), TPB, 0, stream>>>(h1, deg, b1, out1, n, 64);
    k_aggregate<<<blocks((long long)E * 16), TPB, 0, stream>>>(h1, src, dst, deg, out1, E, 64, 16);
    k_relu     <<<blocks((long long)n * 64), TPB, 0, stream>>>(out1, (long long)n * 64);

    // ---- layer 2: h2 = relu(out1) @ W2 ----
    int t2 = mt * 3;                  // N=40 padded to 48 -> 3 col tiles, Npad=48
    k_gemm_wmma_f32<<<(t2 + 7) / 8, TPB, 0, stream>>>(out1, W2, h2, 64, 40, 48, 3, t2);
    k_init_out <<<blocks((long long)n * 40), TPB, 0, stream>>>(h2, deg, b2, out, n, 40);
    k_aggregate<<<blocks((long long)E * 10), TPB, 0, stream>>>(h2, src, dst, deg, out, E, 40, 10);

    // ---- log_softmax ----
    k_logsoftmax<<<blocks((long long)n * 32), TPB, 0, stream>>>(out, n);
}